// PeriodicRadiusGraph_44659069944226
// MI455X (gfx1250) — compile-verified
//
#include <hip/hip_runtime.h>
#include <math.h>

typedef float v2f __attribute__((ext_vector_type(2)));
typedef float v8f __attribute__((ext_vector_type(8)));

#define N_ATOMS 1024
#define CUTOFF  5.0f

// One wave (32 threads) computes a 16x16 pair tile for all 27 images.
// D = A x B + C with V_WMMA_F32_16X16X4_F32 gives d^2 for the whole tile:
//   A row i = [rx, ry, rz, 1]      (r = centered cartesian coords)
//   B col j = [-2sx, -2sy, -2sz, |s|^2],  s = r_j + image_offset_cart
//   C row i = |r_i|^2
// Elements possibly in range (d2 < 25.05, ~0.1%) are recomputed exactly.
__global__ __launch_bounds__(32)
void PeriodicRadiusGraph_gfx1250_kernel(const float* __restrict__ frac,
                                        const float* __restrict__ cell,
                                        float* __restrict__ out)
{
    const int lane = threadIdx.x;      // 0..31
    const int half = lane >> 4;        // 0: lanes 0-15, 1: lanes 16-31
    const int ln   = lane & 15;
    const int i0 = blockIdx.y * 16;
    const int j0 = blockIdx.x * 16;

    // cell rows into registers (row-vector convention: vec_e = sum_d f_d * cell[d][e])
    const float c00 = cell[0], c01 = cell[1], c02 = cell[2];
    const float c10 = cell[3], c11 = cell[4], c12 = cell[5];
    const float c20 = cell[6], c21 = cell[7], c22 = cell[8];

    // centered cartesian: r_e = sum_d (f_d - 0.5) * cell[d][e]  (translation-invariant d2)
    auto cart = [&](float f0, float f1, float f2, float& r0, float& r1, float& r2) {
        f0 -= 0.5f; f1 -= 0.5f; f2 -= 0.5f;
        r0 = f0 * c00 + f1 * c10 + f2 * c20;
        r1 = f0 * c01 + f1 * c11 + f2 * c21;
        r2 = f0 * c02 + f1 * c12 + f2 * c22;
    };

    // ---- per-lane j column (fixed for whole tile) ----
    const int j = j0 + ln;
    const float fj0 = frac[j * 3 + 0], fj1 = frac[j * 3 + 1], fj2 = frac[j * 3 + 2];
    float rj0, rj1, rj2;
    cart(fj0, fj1, fj2, rj0, rj1, rj2);

    // ---- A operand (16x4 f32): row M = ln; VGPR0={K0|K2}, VGPR1={K1|K3} ----
    const int ia = i0 + ln;
    float ra0, ra1, ra2;
    cart(frac[ia * 3 + 0], frac[ia * 3 + 1], frac[ia * 3 + 2], ra0, ra1, ra2);
    v2f A;
    A.x = half ? ra2 : ra0;   // K2 : K0
    A.y = half ? 1.0f : ra1;  // K3 : K1

    // ---- C operand (16x16 f32): VGPR v -> row M = v + half*8; value |r_i|^2 ----
    v8f C;
#pragma unroll
    for (int v = 0; v < 8; ++v) {
        const int ic = i0 + v + half * 8;
        float r0, r1, r2;
        cart(frac[ic * 3 + 0], frac[ic * 3 + 1], frac[ic * 3 + 2], r0, r1, r2);
        C[v] = r0 * r0 + r1 * r1 + r2 * r2;
    }

    // tile staging: [i_local][j_local*27 + k], 16*16*27 floats = 27.6 KB
    __shared__ __align__(16) float sT[16 * 16 * 27];

    for (int k = 0; k < 27; ++k) {
        // integer image offset (matches meshgrid 'ij' order: k = a*9 + b*3 + c)
        const float g0 = (float)(k / 9) - 1.0f;
        const float g1 = (float)((k / 3) % 3) - 1.0f;
        const float g2 = (float)(k % 3) - 1.0f;
        // cartesian image offset
        const float o0 = g0 * c00 + g1 * c10 + g2 * c20;
        const float o1 = g0 * c01 + g1 * c11 + g2 * c21;
        const float o2 = g0 * c02 + g1 * c12 + g2 * c22;
        const float s0 = rj0 + o0, s1 = rj1 + o1, s2 = rj2 + o2;
        const float ss = s0 * s0 + s1 * s1 + s2 * s2;

        // B operand (4x16 f32): col N = ln; VGPR0={K0|K2}, VGPR1={K1|K3}
        v2f B;
        B.x = half ? (-2.0f * s2) : (-2.0f * s0);
        B.y = half ? ss : (-2.0f * s1);

        // D[i][j] = |r_i|^2 - 2 r_i . s_jk + |s_jk|^2 = d^2(i, j, k)
        v8f D = __builtin_amdgcn_wmma_f32_16x16x4_f32(
            /*neg_a=*/false, A, /*neg_b=*/false, B,
            /*c_mod=*/(short)0, C, /*reuse_a=*/false, /*reuse_b=*/false);

#pragma unroll
        for (int v = 0; v < 8; ++v) {
            const int il = v + half * 8;           // row M in tile
            const float d2a = D[v];
            float outv = 0.0f;
            if (d2a < 25.05f) {                    // cutoff^2 + cancellation slack
                // exact recompute, same op order as reference: (f_j - f_i + off) @ cell
                const int i = i0 + il;
                const float df0 = fj0 - frac[i * 3 + 0] + g0;
                const float df1 = fj1 - frac[i * 3 + 1] + g1;
                const float df2 = fj2 - frac[i * 3 + 2] + g2;
                const float v0 = df0 * c00 + df1 * c10 + df2 * c20;
                const float v1 = df0 * c01 + df1 * c11 + df2 * c21;
                const float v2 = df0 * c02 + df1 * c12 + df2 * c22;
                const float d2 = v0 * v0 + v1 * v1 + v2 * v2;
                const float dist = sqrtf(d2);
                outv = (d2 > 1e-12f && dist < CUTOFF) ? dist : 0.0f;
            }
            sT[(il * 16 + ln) * 27 + k] = outv;
        }
    }
    __syncthreads();

    // ---- coalesced writeout: each output row i is 16*27 = 432 contiguous floats ----
    // 432 floats = 108 float4; base offset 27*(i*1024+j0) is a multiple of 432 -> 16B aligned
#pragma unroll 1
    for (int il = 0; il < 16; ++il) {
        const size_t gbase = ((size_t)(i0 + il) * N_ATOMS + j0) * 27;
        const float4* src = (const float4*)&sT[il * 432];
        float4* dst = (float4*)(out + gbase);
        for (int t = lane; t < 108; t += 32) {
            dst[t] = src[t];
        }
    }
}

extern "C" void kernel_launch(void* const* d_in, const int* in_sizes, int n_in,
                              void* d_out, int out_size, void* d_ws, size_t ws_size,
                              hipStream_t stream) {
    (void)in_sizes; (void)n_in; (void)d_ws; (void)ws_size; (void)out_size;
    const float* frac = (const float*)d_in[0];   // [1024, 3] fp32
    const float* cell = (const float*)d_in[1];   // [3, 3]    fp32
    float* out = (float*)d_out;                  // [1024, 1024, 27] fp32

    dim3 grid(N_ATOMS / 16, N_ATOMS / 16);       // 64 x 64 tiles
    PeriodicRadiusGraph_gfx1250_kernel<<<grid, 32, 0, stream>>>(frac, cell, out);
}